// GAT_22084721836315
// MI455X (gfx1250) — compile-verified
//
#include <hip/hip_runtime.h>
#include <cmath>
#include <cstddef>

// GAT (2-layer, PyG GATConv semantics) for MI455X / gfx1250, wave32.
// Dense projections use V_WMMA_F32_16X16X4_F32; sparse attention is
// L2-resident atomic gather/scatter (node state << 192MB L2).

typedef __attribute__((ext_vector_type(2))) float v2f;
typedef __attribute__((ext_vector_type(8))) float v8f;

#define SLOPE 0.2f

__device__ __forceinline__ void atomic_max_f32(float* addr, float val) {
  // sign-aware int trick -> native global_atomic_max_i32 / min_u32
  if (val >= 0.0f) {
    atomicMax(reinterpret_cast<int*>(addr), __float_as_int(val));
  } else {
    atomicMin(reinterpret_cast<unsigned int*>(addr),
              static_cast<unsigned int>(__float_as_int(val)));
  }
}

// ----------------------------- init kernels -----------------------------
__global__ void fill_f32(float* __restrict__ p, long n, float v) {
  long i = (long)blockIdx.x * blockDim.x + threadIdx.x;
  if (i < n) p[i] = v;
}

__global__ void bias_init(float* __restrict__ p, const float* __restrict__ b,
                          long rows, int cols) {
  long i = (long)blockIdx.x * blockDim.x + threadIdx.x;
  if (i < rows * (long)cols) p[i] = b[i % cols];
}

// ------------------- WMMA GEMM: Hout = act(A[M,K]) @ W[K,NN] -------------------
// One wave32 per 16x16 output tile. W column-tile staged through LDS.
// Fragment layouts per CDNA5 ISA 7.12.2:
//   A 16x4 f32 : lane = M%16 (+16 for K-half 2..3), v[0..1] = K, K+1
//   B 4x16 f32 : lane = N%16 (+16 for K-half 2..3), v[0..1] = K, K+1
//   C 16x16 f32: lanes 0-15 N=lane M=vgpr, lanes 16-31 N=lane-16 M=vgpr+8
template <int K, int NN, bool RELU_IN>
__global__ void __launch_bounds__(32)
gemm_wmma(const float* __restrict__ A, const float* __restrict__ W,
          float* __restrict__ Hout, int M) {
  __shared__ float Bs[K * 16];
  const int lane  = threadIdx.x;
  const int tileM = blockIdx.x;
  const int tileN = blockIdx.y;

  for (int i = lane; i < K * 16; i += 32) {
    int k = i >> 4, n = i & 15;
    Bs[i] = W[(size_t)k * NN + tileN * 16 + n];
  }
  __syncthreads();

  const int mrow  = tileM * 16 + (lane & 15);
  const int row   = mrow < M ? mrow : M - 1;  // clamp: keep EXEC all-ones for WMMA
  const int khalf = (lane >> 4) * 2;          // 0 or 2
  const int ncol  = lane & 15;

  v8f acc = {};
  for (int k = 0; k < K; k += 4) {
    float a0 = A[(size_t)row * K + k + khalf];
    float a1 = A[(size_t)row * K + k + khalf + 1];
    if (RELU_IN) { a0 = fmaxf(a0, 0.0f); a1 = fmaxf(a1, 0.0f); }
    v2f av = {a0, a1};
    v2f bv = {Bs[(k + khalf) * 16 + ncol], Bs[(k + khalf + 1) * 16 + ncol]};
    acc = __builtin_amdgcn_wmma_f32_16x16x4_f32(false, av, false, bv,
                                                (short)0, acc, false, false);
  }

  const int col   = tileN * 16 + ncol;
  const int rbase = (lane >> 4) * 8;
#pragma unroll
  for (int i = 0; i < 8; ++i) {
    int r = tileM * 16 + rbase + i;
    if (r < M) Hout[(size_t)r * NN + col] = acc[i];
  }
}

// ------------------- attention scalars: al = sum_c h[n,h,c]*a[h,c] -------------------
template <int HH, int CC>
__global__ void attn_scalars(const float* __restrict__ Hm,
                             const float* __restrict__ a_s,
                             const float* __restrict__ a_d,
                             float* __restrict__ als, float* __restrict__ ald,
                             int n) {
  int idx = blockIdx.x * blockDim.x + threadIdx.x;  // node*HH + head
  if (idx >= n * HH) return;
  int node = idx / HH, head = idx % HH;
  const float* hp = Hm + (size_t)node * HH * CC + (size_t)head * CC;
  float ss = 0.f, sd = 0.f;
#pragma unroll
  for (int c = 0; c < CC; ++c) {
    float v = hp[c];
    ss += v * a_s[head * CC + c];
    sd += v * a_d[head * CC + c];
  }
  als[idx] = ss;
  ald[idx] = sd;
}

// ------------------- edge pass 1: leaky-relu logits + segment max -------------------
template <int HH>
__global__ void edge_logits_max(const int* __restrict__ ei, int E, int N,
                                const float* __restrict__ als,
                                const float* __restrict__ ald,
                                float* __restrict__ ebuf,
                                float* __restrict__ mbuf) {
  long idx = (long)blockIdx.x * blockDim.x + threadIdx.x;
  long tot = (long)(E + N) * HH;
  if (idx >= tot) return;
  int e = (int)(idx / HH), h = (int)(idx % HH);
  int s = (e < E) ? ei[e] : (e - E);          // self loops appended
  int d = (e < E) ? ei[E + e] : (e - E);
  float v = als[s * HH + h] + ald[d * HH + h];
  v = (v > 0.f) ? v : SLOPE * v;
  ebuf[idx] = v;
  atomic_max_f32(&mbuf[d * HH + h], v);
}

// ------------------- edge pass 2: exp(e - m[dst]) + segment sum -------------------
template <int HH>
__global__ void edge_exp_sum(const int* __restrict__ ei, int E, int N,
                             const float* __restrict__ mbuf,
                             float* __restrict__ ebuf,
                             float* __restrict__ zbuf) {
  long idx = (long)blockIdx.x * blockDim.x + threadIdx.x;
  long tot = (long)(E + N) * HH;
  if (idx >= tot) return;
  int e = (int)(idx / HH), h = (int)(idx % HH);
  int d = (e < E) ? ei[E + e] : (e - E);
  float v = expf(ebuf[idx] - mbuf[d * HH + h]);
  ebuf[idx] = v;
  atomicAdd(&zbuf[d * HH + h], v);
}

// ------------------- edge pass 3: out[dst] += alpha * h[src] -------------------
template <int HH, int CC>
__global__ void edge_aggregate(const int* __restrict__ ei, int E, int N,
                               const float* __restrict__ ebuf,
                               const float* __restrict__ zbuf,
                               const float* __restrict__ Hm,
                               float* __restrict__ out) {
  long idx = (long)blockIdx.x * blockDim.x + threadIdx.x;
  long tot = (long)(E + N) * HH;
  if (idx >= tot) return;
  int e = (int)(idx / HH), h = (int)(idx % HH);
  int s = (e < E) ? ei[e] : (e - E);
  int d = (e < E) ? ei[E + e] : (e - E);
  float alpha = ebuf[idx] / (zbuf[d * HH + h] + 1e-16f);
  const float* hp = Hm + (size_t)s * HH * CC + (size_t)h * CC;
  float* op = out + (size_t)d * HH * CC + (size_t)h * CC;
#pragma unroll 4
  for (int c = 0; c < CC; ++c) atomicAdd(&op[c], hp[c] * alpha);
}

// ----------------------------- launch -----------------------------
extern "C" void kernel_launch(void* const* d_in, const int* in_sizes, int n_in,
                              void* d_out, int out_size, void* d_ws, size_t ws_size,
                              hipStream_t stream) {
  (void)n_in; (void)out_size; (void)ws_size;
  const float* x    = (const float*)d_in[0];
  const int*   ei   = (const int*)d_in[1];   // (2,E) int32
  // d_in[2] = edge_attr, ignored (edge_dim=None)
  const float* W1   = (const float*)d_in[3];
  const float* a_s1 = (const float*)d_in[4];
  const float* a_d1 = (const float*)d_in[5];
  const float* b1   = (const float*)d_in[6];
  const float* W2   = (const float*)d_in[7];
  const float* a_s2 = (const float*)d_in[8];
  const float* a_d2 = (const float*)d_in[9];
  const float* b2   = (const float*)d_in[10];
  float* out = (float*)d_out;

  const int  N    = in_sizes[0] / 128;
  const int  E    = in_sizes[1] / 2;
  const long Etot = (long)E + N;

  // workspace layout (fp32), ~82 MB total
  float* ws   = (float*)d_ws;
  float* h1   = ws;  ws += (size_t)N * 128;  // layer1 projected features
  float* o1   = ws;  ws += (size_t)N * 128;  // layer1 aggregated output
  float* h2   = ws;  ws += (size_t)N * 64;   // layer2 projected features
  float* als1 = ws;  ws += (size_t)N * 4;
  float* ald1 = ws;  ws += (size_t)N * 4;
  float* m1   = ws;  ws += (size_t)N * 4;
  float* z1   = ws;  ws += (size_t)N * 4;
  float* als2 = ws;  ws += (size_t)N;
  float* ald2 = ws;  ws += (size_t)N;
  float* m2   = ws;  ws += (size_t)N;
  float* z2   = ws;  ws += (size_t)N;
  float* ebuf = ws;                          // Etot*4 (reused Etot*1 in layer2)

  const int TB = 256;
  auto nb = [](long n) { return (unsigned)((n + 255) / 256); };

  // ---------------- layer 1 (H=4, C=32, concat) ----------------
  fill_f32<<<nb((long)N * 4), TB, 0, stream>>>(m1, (long)N * 4, -INFINITY);
  fill_f32<<<nb((long)N * 4), TB, 0, stream>>>(z1, (long)N * 4, 0.f);
  bias_init<<<nb((long)N * 128), TB, 0, stream>>>(o1, b1, N, 128);

  dim3 g1((N + 15) / 16, 128 / 16);
  gemm_wmma<128, 128, false><<<g1, 32, 0, stream>>>(x, W1, h1, N);
  attn_scalars<4, 32><<<nb((long)N * 4), TB, 0, stream>>>(h1, a_s1, a_d1, als1, ald1, N);
  edge_logits_max<4><<<nb(Etot * 4), TB, 0, stream>>>(ei, E, N, als1, ald1, ebuf, m1);
  edge_exp_sum<4><<<nb(Etot * 4), TB, 0, stream>>>(ei, E, N, m1, ebuf, z1);
  edge_aggregate<4, 32><<<nb(Etot * 4), TB, 0, stream>>>(ei, E, N, ebuf, z1, h1, o1);

  // ---------------- layer 2 (H=1, C=64, mean==identity) ----------------
  fill_f32<<<nb(N), TB, 0, stream>>>(m2, N, -INFINITY);
  fill_f32<<<nb(N), TB, 0, stream>>>(z2, N, 0.f);
  bias_init<<<nb((long)N * 64), TB, 0, stream>>>(out, b2, N, 64);

  dim3 g2((N + 15) / 16, 64 / 16);
  gemm_wmma<128, 64, true><<<g2, 32, 0, stream>>>(o1, W2, h2, N);  // ReLU fused on A-load
  attn_scalars<1, 64><<<nb(N), TB, 0, stream>>>(h2, a_s2, a_d2, als2, ald2, N);
  edge_logits_max<1><<<nb(Etot), TB, 0, stream>>>(ei, E, N, als2, ald2, ebuf, m2);
  edge_exp_sum<1><<<nb(Etot), TB, 0, stream>>>(ei, E, N, m2, ebuf, z2);
  edge_aggregate<1, 64><<<nb(Etot), TB, 0, stream>>>(ei, E, N, ebuf, z2, h2, out);
}